// V2PModule_26259430048621
// MI455X (gfx1250) — compile-verified
//
#include <hip/hip_runtime.h>
#include <hip/hip_bf16.h>

typedef __attribute__((ext_vector_type(2))) float v2f;
typedef __attribute__((ext_vector_type(8))) float v8f;

#define C_DIM 64
#define NSAMP 16

// ---------------------------------------------------------------------------
// Phase 1: per-voxel  Hv[m,o] = relu( (sum_c vf[m,c]*W[o,c] - mean[o]) *
//                                     gamma[o]/sqrt(var[o]+eps) + beta[o] )
// One wave handles 16 voxel rows x all 64 output channels using
// V_WMMA_F32_16X16X4_F32 (fp32 A/B, 16x16 f32 accumulator, K=4 per step).
//   A (16x4):  lanes 0-15 -> rows 0..15, vgpr0/1 = K {k0,k0+1}; lanes 16-31
//              same rows, K {k0+2,k0+3}.
//   B (4x16):  B[k][col] = W[col_global*64 + k]; lane = (k>=k0+2?16:0)+col.
//   C (16x16): vgpr r -> row r (lanes 0-15) / row r+8 (lanes 16-31),
//              column = lane & 15.
// ---------------------------------------------------------------------------
__global__ __launch_bounds__(256) void v2p_gemm_bn_relu(
    const float* __restrict__ vf, const float* __restrict__ W,
    const float* __restrict__ gamma, const float* __restrict__ beta,
    const float* __restrict__ bn_mean, const float* __restrict__ bn_var,
    float* __restrict__ Hv, int M)
{
    const int wave = threadIdx.x >> 5;
    const int lane = threadIdx.x & 31;
    const int tile = blockIdx.x * 8 + wave;          // 16 rows per tile
    const int row0 = tile * 16;
    if (row0 >= M) return;                            // wave-uniform exit

    const int mrow  = lane & 15;                      // row within tile / B col
    const int khalf = (lane >> 4) << 1;               // 0 or 2
    int arow = row0 + mrow;
    if (arow >= M) arow = M - 1;                      // clamp partial tile loads

    const float* __restrict__ aptr = vf + (size_t)arow * C_DIM;

    v8f acc[4] = {};                                  // o-tiles 0..3 (o = t*16+col)

    #pragma unroll
    for (int k0 = 0; k0 < C_DIM; k0 += 4) {
        const int kk = k0 + khalf;
        v2f a;
        a.x = aptr[kk];
        a.y = aptr[kk + 1];
        #pragma unroll
        for (int t = 0; t < 4; ++t) {
            v2f b;
            const int og = t * 16 + mrow;             // global output channel
            b.x = W[og * C_DIM + kk];
            b.y = W[og * C_DIM + kk + 1];
            acc[t] = __builtin_amdgcn_wmma_f32_16x16x4_f32(
                /*neg_a=*/false, a, /*neg_b=*/false, b,
                /*c_mod=*/(short)0, acc[t],
                /*reuse_a=*/false, /*reuse_b=*/false);
        }
    }

    // Fused BN + ReLU epilogue, then scatter 16x64 tile to Hv.
    const int rowadd = (lane >> 4) << 3;              // 0 or 8
    #pragma unroll
    for (int t = 0; t < 4; ++t) {
        const int o  = t * 16 + mrow;
        const float sc = gamma[o] * rsqrtf(bn_var[o] + 1e-5f);
        const float mn = bn_mean[o];
        const float bt = beta[o];
        #pragma unroll
        for (int r = 0; r < 8; ++r) {
            const int row = row0 + r + rowadd;
            if (row < M) {
                float v = (acc[t][r] - mn) * sc + bt;
                Hv[(size_t)row * C_DIM + o] = fmaxf(v, 0.0f);
            }
        }
    }
}

// ---------------------------------------------------------------------------
// Phase 2: masked geometric pooling.
// One 64-thread block per point; thread o = output channel.  All per-sample
// metadata (p_map, p_mask, v_indices row) is wave-uniform -> scalar loads;
// only the Hv row gather is a vector load (256B contiguous, fully coalesced).
// Hv (25.6 MB) is L2-resident on MI455X (192 MB L2).
// ---------------------------------------------------------------------------
__global__ __launch_bounds__(64) void v2p_pool(
    const int*   __restrict__ v_indices,
    const float* __restrict__ p_coords,
    const float* __restrict__ p_features,
    const int*   __restrict__ p_map,
    const float* __restrict__ p_mask,
    const float* __restrict__ Hv,
    float* __restrict__ out, int N)
{
    const int n = blockIdx.x;
    if (n >= N) return;
    const int o = threadIdx.x;                        // 0..63
    const int comp = o & 3;                           // geo component o%4

    // centers = p_coords[:, 1:4]
    const float cx = p_coords[n * 4 + 1];
    const float cy = p_coords[n * 4 + 2];
    const float cz = p_coords[n * 4 + 3];

    // 1 / NORMALIZER = {10, 10, 5, 1/0.06}
    const float invn = (comp == 0) ? 10.0f :
                       (comp == 1) ? 10.0f :
                       (comp == 2) ? 5.0f  : (1.0f / 0.06f);

    float acc  = 0.0f;
    float msum = 0.0f;

    int m_next = p_map[n * NSAMP];
    #pragma unroll
    for (int s = 0; s < NSAMP; ++s) {
        const int m = m_next;
        if (s + 1 < NSAMP) {
            m_next = p_map[n * NSAMP + s + 1];
            __builtin_prefetch(Hv + (size_t)m_next * C_DIM + o, 0, 3);
        }
        const float mask = p_mask[n * NSAMP + s];

        // v_xyz = (v_indices[m][3,2,1] + 0.5) * V_SIZE + PC_MIN
        const int i1 = v_indices[m * 4 + 1];
        const int i2 = v_indices[m * 4 + 2];
        const int i3 = v_indices[m * 4 + 3];
        const float gx = ((float)i3 + 0.5f) * 0.1f +   0.0f - cx;
        const float gy = ((float)i2 + 0.5f) * 0.1f + (-40.0f) - cy;
        const float gz = ((float)i1 + 0.5f) * 0.2f + ( -3.0f) - cz;
        const float dist = gx * gx + gy * gy + gz * gz;

        const float geo = ((comp == 0) ? gx :
                           (comp == 1) ? gy :
                           (comp == 2) ? gz : dist) * invn;

        const float h = Hv[(size_t)m * C_DIM + o];
        acc  = fmaf(h * geo, mask, acc);
        msum += mask;
    }

    const float denom = fmaxf(msum, 1.0f);
    out[(size_t)n * C_DIM + o] = p_features[(size_t)n * C_DIM + o] + acc / denom;
}

// ---------------------------------------------------------------------------
extern "C" void kernel_launch(void* const* d_in, const int* in_sizes, int n_in,
                              void* d_out, int out_size, void* d_ws, size_t ws_size,
                              hipStream_t stream)
{
    const int*   v_indices  = (const int*)  d_in[0];
    const float* v_features = (const float*)d_in[1];
    const float* p_coords   = (const float*)d_in[2];
    const float* p_features = (const float*)d_in[3];
    const int*   p_map      = (const int*)  d_in[4];
    const float* p_mask     = (const float*)d_in[5];
    const float* W          = (const float*)d_in[6];
    const float* gamma      = (const float*)d_in[7];
    const float* beta       = (const float*)d_in[8];
    const float* bn_mean    = (const float*)d_in[9];
    const float* bn_var     = (const float*)d_in[10];
    float* out = (float*)d_out;

    const int M = in_sizes[1] / C_DIM;   // 100000
    const int N = in_sizes[3] / C_DIM;   // 131072

    float* Hv = (float*)d_ws;            // M*64*4 = 25.6 MB scratch

    const int tiles = (M + 15) / 16;
    dim3 g1((tiles + 7) / 8), b1(256);
    v2p_gemm_bn_relu<<<g1, b1, 0, stream>>>(v_features, W, gamma, beta,
                                            bn_mean, bn_var, Hv, M);

    dim3 g2(N), b2(C_DIM);
    v2p_pool<<<g2, b2, 0, stream>>>(v_indices, p_coords, p_features,
                                    p_map, p_mask, Hv, out, N);
}